// MultiHeadSelfAttention_88725434400988
// MI455X (gfx1250) — compile-verified
//
#include <hip/hip_runtime.h>

typedef float v2f __attribute__((ext_vector_type(2)));
typedef float v4f __attribute__((ext_vector_type(4)));
typedef float v8f __attribute__((ext_vector_type(8)));
typedef unsigned int u32;
typedef u32 u32x4 __attribute__((ext_vector_type(4)));
typedef int i32x4 __attribute__((ext_vector_type(4)));
typedef int i32x8 __attribute__((ext_vector_type(8)));

// D = A(16x4) * B(4x16) + C, fp32, wave32.
__device__ __forceinline__ v8f wmma4(v2f a, v2f b, v8f c) {
  return __builtin_amdgcn_wmma_f32_16x16x4_f32(
      /*neg_a=*/false, a, /*neg_b=*/false, b,
      /*c_mod=*/(short)0, c, /*reuse_a=*/false, /*reuse_b=*/false);
}

#if __has_builtin(__builtin_amdgcn_tensor_load_to_lds) && \
    __has_builtin(__builtin_amdgcn_s_wait_tensorcnt)
#define USE_TDM 1
#else
#define USE_TDM 0
#endif

#if USE_TDM
// 2-D tile DMA: global (row-major, stride_d0 elems/row) -> LDS, fp32 elems,
// optional LDS row padding (pad codes per CDNA5 ISA 8.4: interval 1<<(c+1)
// dwords, amount c+1 dwords).  6-arg builtin (clang-23 / therock headers):
// (g0 x4, g1 x8, g2 x4, g3 x4, x8, cpol).
__device__ __forceinline__ void tdm_load_2d(u32 lds_byte_off, const void* gptr,
                                            u32 tensor_d0, u32 tensor_d1,
                                            u32 stride_d0, u32 tile_d0,
                                            u32 tile_d1, u32 pad_int_code,
                                            u32 pad_amt_code, bool pad_en) {
  unsigned long long ga = (unsigned long long)(uintptr_t)gptr;
  u32x4 g0;
  g0[0] = 1u;                                             // count=1, user D#
  g0[1] = lds_byte_off;                                   // lds_addr
  g0[2] = (u32)ga;                                        // global_addr[31:0]
  g0[3] = (u32)((ga >> 32) & 0x1FFFFFFull) | (2u << 30);  // addr[56:32]|type=2
  i32x8 g1;
  g1[0] = (int)((2u << 16) |                               // data_size = 4B
                (pad_en ? (1u << 20) : 0u) | (pad_int_code << 22) |
                (pad_amt_code << 25));
  g1[1] = (int)((tensor_d0 & 0xFFFFu) << 16);              // tensor_dim0 lo16
  g1[2] = (int)((tensor_d0 >> 16) | ((tensor_d1 & 0xFFFFu) << 16));
  g1[3] = (int)((tensor_d1 >> 16) | (tile_d0 << 16));      // | tile_dim0
  g1[4] = (int)tile_d1;                                    // tile_dim2 = 0
  g1[5] = (int)stride_d0;                                  // dim0 stride lo32
  g1[6] = 0;                                               // stride hi, dim1 s
  g1[7] = 0;
  i32x4 z4 = {0, 0, 0, 0};       // groups 2/3 unused for 2-D tiles
  i32x8 z8 = {0, 0, 0, 0, 0, 0, 0, 0};
  __builtin_amdgcn_tensor_load_to_lds(g0, g1, z4, z4, z8, 0);
}
#endif

// ---------------------------------------------------------------------------
// Problem dims
// ---------------------------------------------------------------------------
#define G1_M 2048  // B*S
#define G1_K 2048  // F*E
#define G1_N 4096  // A*F*NH

// ---------------------------------------------------------------------------
// K1: Q = Hs @ Wq, K = Hs @ Wk  (blockIdx.z selects)
//   WG = 128 thr (4 waves), tile 64x128, wave = 32x64, BK = 16.
//   Double-buffered TDM tile loads (TENSORcnt) overlapped with WMMA.
// ---------------------------------------------------------------------------
#define AP1 20   // As pitch: TDM pad 16dw + 4dw; banks 20i mod 64 distinct
#define BP1 136  // Bs pitch: TDM pad 128dw + 8dw; 2*136 mod 64 = 16
#define K1_AN (64 * AP1)
#define K1_BN (16 * BP1)

__global__ __launch_bounds__(128) void qk_proj_kernel(
    const float* __restrict__ Hs, const float* __restrict__ Wq,
    const float* __restrict__ Wk, float* __restrict__ q_ws,
    float* __restrict__ k_ws) {
  __shared__ float As[2][K1_AN];
  __shared__ float Bs[2][K1_BN];

  const float* W = blockIdx.z ? Wk : Wq;
  float* C = blockIdx.z ? k_ws : q_ws;
  const int nblk = blockIdx.x * 128;
  const int mblk = blockIdx.y * 64;
  const int tid = threadIdx.x;
  const int lane = tid & 31, wave = tid >> 5;
  const int half = lane >> 4, l16 = lane & 15;
  const int wm = (wave & 1) * 32;
  const int wn = (wave >> 1) * 64;

  v8f acc[2][4];
#pragma unroll
  for (int i = 0; i < 2; i++)
#pragma unroll
    for (int j = 0; j < 4; j++)
#pragma unroll
      for (int e = 0; e < 8; e++) acc[i][j][e] = 0.0f;

#if USE_TDM
  const u32 lds_a0 = (u32)(uintptr_t)&As[0][0];
  const u32 lds_a1 = (u32)(uintptr_t)&As[1][0];
  const u32 lds_b0 = (u32)(uintptr_t)&Bs[0][0];
  const u32 lds_b1 = (u32)(uintptr_t)&Bs[1][0];
  if (wave == 0) {  // prologue: fill buffer 0 (k0 = 0)
    tdm_load_2d(lds_a0, Hs + (size_t)mblk * G1_K, G1_K, G1_M, G1_K, 16, 64,
                /*pad 16dw->+4dw*/ 3, 3, true);
    tdm_load_2d(lds_b0, W + nblk, G1_N, G1_K, G1_N, 128, 16,
                /*pad 128dw->+8dw*/ 6, 7, true);
  }
  for (int k0 = 0; k0 < G1_K; k0 += 16) {
    const int buf = (k0 >> 4) & 1;
    if (wave == 0) {
      if (k0 + 16 < G1_K) {  // issue next slab into other buffer, then wait
        tdm_load_2d(buf ? lds_a0 : lds_a1,
                    Hs + (size_t)mblk * G1_K + k0 + 16, G1_K, G1_M, G1_K, 16,
                    64, 3, 3, true);
        tdm_load_2d(buf ? lds_b0 : lds_b1,
                    W + (size_t)(k0 + 16) * G1_N + nblk, G1_N, G1_K, G1_N, 128,
                    16, 6, 7, true);
        __builtin_amdgcn_s_wait_tensorcnt(2);  // oldest 2 (current buf) done
      } else {
        __builtin_amdgcn_s_wait_tensorcnt(0);
      }
    }
    __syncthreads();
#pragma unroll
    for (int kk = 0; kk < 16; kk += 4) {
      v2f a0 = *(const v2f*)&As[buf][(wm + l16) * AP1 + kk + 2 * half];
      v2f a1 = *(const v2f*)&As[buf][(wm + 16 + l16) * AP1 + kk + 2 * half];
#pragma unroll
      for (int j = 0; j < 4; j++) {
        const int nc = wn + j * 16 + l16;
        v2f b;
        b.x = Bs[buf][(kk + 2 * half) * BP1 + nc];
        b.y = Bs[buf][(kk + 2 * half + 1) * BP1 + nc];
        acc[0][j] = wmma4(a0, b, acc[0][j]);
        acc[1][j] = wmma4(a1, b, acc[1][j]);
      }
    }
    __syncthreads();  // readers done before next issue overwrites this buf
  }
#else  // fallback: direct copies, single buffer
  const int a_row = tid >> 2, a_col = (tid & 3) * 4;
  for (int k0 = 0; k0 < G1_K; k0 += 16) {
    {
      const float* src = Hs + (size_t)(mblk + a_row) * G1_K + k0 + a_col;
      *(v4f*)&As[0][a_row * AP1 + a_col] = *(const v4f*)src;
      *(v4f*)&As[0][(a_row + 32) * AP1 + a_col] = *(const v4f*)(src + 32 * G1_K);
    }
#pragma unroll
    for (int i = 0; i < 4; i++) {
      int idx = i * 128 + tid;
      int r = idx >> 5, c = (idx & 31) * 4;
      *(v4f*)&Bs[0][r * BP1 + c] =
          *(const v4f*)&W[(size_t)(k0 + r) * G1_N + nblk + c];
    }
    __syncthreads();
#pragma unroll
    for (int kk = 0; kk < 16; kk += 4) {
      v2f a0 = *(const v2f*)&As[0][(wm + l16) * AP1 + kk + 2 * half];
      v2f a1 = *(const v2f*)&As[0][(wm + 16 + l16) * AP1 + kk + 2 * half];
#pragma unroll
      for (int j = 0; j < 4; j++) {
        const int nc = wn + j * 16 + l16;
        v2f b;
        b.x = Bs[0][(kk + 2 * half) * BP1 + nc];
        b.y = Bs[0][(kk + 2 * half + 1) * BP1 + nc];
        acc[0][j] = wmma4(a0, b, acc[0][j]);
        acc[1][j] = wmma4(a1, b, acc[1][j]);
      }
    }
    __syncthreads();
  }
#endif

#pragma unroll
  for (int i = 0; i < 2; i++)
#pragma unroll
    for (int j = 0; j < 4; j++) {
      const int rbase = mblk + wm + i * 16 + half * 8;
      const int c = nblk + wn + j * 16 + l16;
#pragma unroll
      for (int r = 0; r < 8; r++)
        C[(size_t)(rbase + r) * G1_N + c] = acc[i][j][r];
    }
}

// ---------------------------------------------------------------------------
// K2: V = Hs_r(65536x64) @ Wv(64x128).  WG=128 thr, 64 rows x 128 cols.
// ---------------------------------------------------------------------------
#define AP2 68
#define BP2 136

__global__ __launch_bounds__(128) void v_proj_kernel(
    const float* __restrict__ Hs, const float* __restrict__ Wv,
    float* __restrict__ v_ws) {
  __shared__ float As[64 * AP2];
  __shared__ float Bs[64 * BP2];
  const int tid = threadIdx.x, lane = tid & 31, wave = tid >> 5;
  const int half = lane >> 4, l16 = lane & 15;
  const int mblk = blockIdx.x * 64;

#pragma unroll
  for (int i = 0; i < 16; i++) {  // Wv whole: 64x128
    int idx = i * 128 + tid;
    int r = idx >> 5, c = (idx & 31) * 4;
    *(v4f*)&Bs[r * BP2 + c] = *(const v4f*)&Wv[r * 128 + c];
  }
#pragma unroll
  for (int i = 0; i < 8; i++) {  // A tile: 64x64
    int idx = i * 128 + tid;
    int r = idx >> 4, c = (idx & 15) * 4;
    *(v4f*)&As[r * AP2 + c] = *(const v4f*)&Hs[(size_t)(mblk + r) * 64 + c];
  }
  __syncthreads();

  v8f acc[8];
#pragma unroll
  for (int j = 0; j < 8; j++)
#pragma unroll
    for (int e = 0; e < 8; e++) acc[j][e] = 0.0f;
  const int wm = wave * 16;
#pragma unroll
  for (int kk = 0; kk < 64; kk += 4) {
    v2f a = *(const v2f*)&As[(wm + l16) * AP2 + kk + 2 * half];
#pragma unroll
    for (int j = 0; j < 8; j++) {
      v2f b;
      b.x = Bs[(kk + 2 * half) * BP2 + j * 16 + l16];
      b.y = Bs[(kk + 2 * half + 1) * BP2 + j * 16 + l16];
      acc[j] = wmma4(a, b, acc[j]);
    }
  }
#pragma unroll
  for (int j = 0; j < 8; j++) {
    const int rbase = mblk + wm + half * 8;
    const int c = j * 16 + l16;
#pragma unroll
    for (int r = 0; r < 8; r++)
      v_ws[(size_t)(rbase + r) * 128 + c] = acc[j][r];
  }
}

// ---------------------------------------------------------------------------
// K3: attention over flat (512, 256, 64) Q/K/V.
//   WG = 256 thr (8 waves) per (batch, 64-row q-tile).  Scores live in WMMA
//   accumulators; softmax via shfl_xor + LDS combine; P re-staged through LDS
//   (A-matrix layout) for the P@V WMMAs; V streamed in 64-key chunks.
// ---------------------------------------------------------------------------
#define QP3 68
#define KP3 72

__global__ __launch_bounds__(256) void attn_kernel(
    const float* __restrict__ q_ws, const float* __restrict__ k_ws,
    const float* __restrict__ v_ws, float* __restrict__ attn_ws) {
  __shared__ float Qs[64 * QP3];   // Q tile, reused for P chunks
  __shared__ float KVs[64 * KP3];  // K chunks, then V chunks
  __shared__ float RedM[64 * 2];
  __shared__ float RedS[64 * 2];

  const int tid = threadIdx.x, lane = tid & 31, wave = tid >> 5;
  const int half = lane >> 4, l16 = lane & 15;
  const int bh = blockIdx.y, qt = blockIdx.x;
  const size_t base = (size_t)bh * 256 * 64;
  const float* Q = q_ws + base + (size_t)qt * 64 * 64;
  const float* K = k_ws + base;
  const float* V = v_ws + base;

  const int strip = wave & 3;
  const int nh = wave >> 2;

#pragma unroll
  for (int i = 0; i < 4; i++) {
    int idx = i * 256 + tid;
    int r = idx >> 4, c = (idx & 15) * 4;
    *(v4f*)&Qs[r * QP3 + c] = *(const v4f*)&Q[(size_t)r * 64 + c];
  }

  v8f s_acc[4][2];
#pragma unroll
  for (int kc = 0; kc < 4; kc++)
#pragma unroll
    for (int j = 0; j < 2; j++)
#pragma unroll
      for (int e = 0; e < 8; e++) s_acc[kc][j][e] = 0.0f;

  for (int kc = 0; kc < 4; kc++) {  // S = Q @ K^T
    __syncthreads();
#pragma unroll
    for (int i = 0; i < 4; i++) {
      int idx = i * 256 + tid;
      int r = idx >> 4, c = (idx & 15) * 4;
      *(v4f*)&KVs[r * KP3 + c] = *(const v4f*)&K[(size_t)(kc * 64 + r) * 64 + c];
    }
    __syncthreads();
#pragma unroll
    for (int f0 = 0; f0 < 64; f0 += 4) {
      v2f a = *(const v2f*)&Qs[(strip * 16 + l16) * QP3 + f0 + 2 * half];
#pragma unroll
      for (int j = 0; j < 2; j++) {
        const int kb = nh * 32 + j * 16;
        v2f b = *(const v2f*)&KVs[(kb + l16) * KP3 + f0 + 2 * half];
        s_acc[kc][j] = wmma4(a, b, s_acc[kc][j]);
      }
    }
  }

  float rowm[8], rinv[8];
#pragma unroll
  for (int r = 0; r < 8; r++) {
    float m = s_acc[0][0][r];
#pragma unroll
    for (int kc = 0; kc < 4; kc++)
#pragma unroll
      for (int j = 0; j < 2; j++) m = fmaxf(m, s_acc[kc][j][r]);
#pragma unroll
    for (int off = 1; off <= 8; off <<= 1)
      m = fmaxf(m, __shfl_xor(m, off, 32));
    rowm[r] = m;
  }
  if (l16 == 0) {
#pragma unroll
    for (int r = 0; r < 8; r++)
      RedM[(strip * 16 + 8 * half + r) * 2 + nh] = rowm[r];
  }
  __syncthreads();
#pragma unroll
  for (int r = 0; r < 8; r++) {
    const int row = strip * 16 + 8 * half + r;
    rowm[r] = fmaxf(RedM[row * 2], RedM[row * 2 + 1]);
  }
#pragma unroll
  for (int kc = 0; kc < 4; kc++)
#pragma unroll
    for (int j = 0; j < 2; j++)
#pragma unroll
      for (int r = 0; r < 8; r++)
        s_acc[kc][j][r] = __expf(s_acc[kc][j][r] - rowm[r]);
#pragma unroll
  for (int r = 0; r < 8; r++) {
    float s = 0.0f;
#pragma unroll
    for (int kc = 0; kc < 4; kc++)
#pragma unroll
      for (int j = 0; j < 2; j++) s += s_acc[kc][j][r];
#pragma unroll
    for (int off = 1; off <= 8; off <<= 1) s += __shfl_xor(s, off, 32);
    rinv[r] = s;
  }
  if (l16 == 0) {
#pragma unroll
    for (int r = 0; r < 8; r++)
      RedS[(strip * 16 + 8 * half + r) * 2 + nh] = rinv[r];
  }
  __syncthreads();
#pragma unroll
  for (int r = 0; r < 8; r++) {
    const int row = strip * 16 + 8 * half + r;
    rinv[r] = 1.0f / (RedS[row * 2] + RedS[row * 2 + 1]);
  }

  v8f o[2];
#pragma unroll
  for (int j = 0; j < 2; j++)
#pragma unroll
    for (int e = 0; e < 8; e++) o[j][e] = 0.0f;

  for (int kc = 0; kc < 4; kc++) {  // out = P @ V
    __syncthreads();
#pragma unroll
    for (int j = 0; j < 2; j++)
#pragma unroll
      for (int r = 0; r < 8; r++)
        Qs[(strip * 16 + 8 * half + r) * QP3 + nh * 32 + j * 16 + l16] =
            s_acc[kc][j][r] * rinv[r];
#pragma unroll
    for (int i = 0; i < 4; i++) {
      int idx = i * 256 + tid;
      int r = idx >> 4, c = (idx & 15) * 4;
      *(v4f*)&KVs[r * KP3 + c] = *(const v4f*)&V[(size_t)(kc * 64 + r) * 64 + c];
    }
    __syncthreads();
#pragma unroll
    for (int k0 = 0; k0 < 64; k0 += 4) {
      v2f a = *(const v2f*)&Qs[(strip * 16 + l16) * QP3 + k0 + 2 * half];
#pragma unroll
      for (int j = 0; j < 2; j++) {
        const int nb = nh * 32 + j * 16 + l16;
        v2f b;
        b.x = KVs[(k0 + 2 * half) * KP3 + nb];
        b.y = KVs[(k0 + 2 * half + 1) * KP3 + nb];
        o[j] = wmma4(a, b, o[j]);
      }
    }
  }
#pragma unroll
  for (int j = 0; j < 2; j++)
#pragma unroll
    for (int r = 0; r < 8; r++) {
      const int row = qt * 64 + strip * 16 + 8 * half + r;
      const int col = nh * 32 + j * 16 + l16;
      attn_ws[base + (size_t)row * 64 + col] = o[j][r];
    }
}

// ---------------------------------------------------------------------------
// K4: out = relu(attn_r(65536x128) @ Wres_w(128x64) + b).  WG=128 thr.
// ---------------------------------------------------------------------------
#define AP4 68
#define BP4 72

__global__ __launch_bounds__(128) void res_proj_kernel(
    const float* __restrict__ attn_ws, const float* __restrict__ Wres_w,
    const float* __restrict__ Wres_b, float* __restrict__ out) {
  __shared__ float As[64 * AP4];
  __shared__ float Bs[128 * BP4];
  const int tid = threadIdx.x, lane = tid & 31, wave = tid >> 5;
  const int half = lane >> 4, l16 = lane & 15;
  const int mblk = blockIdx.x * 64;

#pragma unroll
  for (int i = 0; i < 16; i++) {
    int idx = i * 128 + tid;
    int r = idx >> 4, c = (idx & 15) * 4;
    *(v4f*)&Bs[r * BP4 + c] = *(const v4f*)&Wres_w[r * 64 + c];
  }
  v8f acc[4];
#pragma unroll
  for (int j = 0; j < 4; j++)
#pragma unroll
    for (int e = 0; e < 8; e++) acc[j][e] = 0.0f;
  const int wm = wave * 16;

  for (int kc = 0; kc < 2; kc++) {
    __syncthreads();
#pragma unroll
    for (int i = 0; i < 8; i++) {
      int idx = i * 128 + tid;
      int r = idx >> 4, c = (idx & 15) * 4;
      *(v4f*)&As[r * AP4 + c] =
          *(const v4f*)&attn_ws[(size_t)(mblk + r) * 128 + kc * 64 + c];
    }
    __syncthreads();
#pragma unroll
    for (int kk = 0; kk < 64; kk += 4) {
      v2f a = *(const v2f*)&As[(wm + l16) * AP4 + kk + 2 * half];
      const int kg = kc * 64 + kk;
#pragma unroll
      for (int j = 0; j < 4; j++) {
        v2f b;
        b.x = Bs[(kg + 2 * half) * BP4 + j * 16 + l16];
        b.y = Bs[(kg + 2 * half + 1) * BP4 + j * 16 + l16];
        acc[j] = wmma4(a, b, acc[j]);
      }
    }
  }
#pragma unroll
  for (int j = 0; j < 4; j++) {
    const int c = j * 16 + l16;
    const float bias = Wres_b[c];
    const int rbase = mblk + wm + 8 * half;
#pragma unroll
    for (int r = 0; r < 8; r++) {
      float v = acc[j][r] + bias;
      out[(size_t)(rbase + r) * 64 + c] = v > 0.0f ? v : 0.0f;
    }
  }
}

// ---------------------------------------------------------------------------
extern "C" void kernel_launch(void* const* d_in, const int* in_sizes, int n_in,
                              void* d_out, int out_size, void* d_ws,
                              size_t ws_size, hipStream_t stream) {
  const float* Hs = (const float*)d_in[0];
  const float* Wq = (const float*)d_in[1];
  const float* Wk = (const float*)d_in[2];
  const float* Wv = (const float*)d_in[3];
  const float* Wres_w = (const float*)d_in[4];
  const float* Wres_b = (const float*)d_in[5];
  float* out = (float*)d_out;

  float* ws = (float*)d_ws;
  const size_t QK = (size_t)G1_M * G1_N;
  float* q_ws = ws;
  float* k_ws = ws + QK;
  float* v_ws = ws + 2 * QK;
  float* a_ws = ws + 3 * QK;

  qk_proj_kernel<<<dim3(G1_N / 128, G1_M / 64, 2), 128, 0, stream>>>(
      Hs, Wq, Wk, q_ws, k_ws);
  v_proj_kernel<<<dim3(65536 / 64), 128, 0, stream>>>(Hs, Wv, v_ws);
  attn_kernel<<<dim3(4, 512), 256, 0, stream>>>(q_ws, k_ws, v_ws, a_ws);
  res_proj_kernel<<<dim3(65536 / 64), 128, 0, stream>>>(a_ws, Wres_w, Wres_b,
                                                        out);
}